// DeformConv2d_21414706938267
// MI455X (gfx1250) — compile-verified
//
#include <hip/hip_runtime.h>

// Problem constants (match reference)
#define NB   4
#define CIN  64
#define COUT 64
#define HH   128
#define WW   128
#define HO_  128
#define WO_  128
#define K2_  9
#define SROW 144   // padded LDS row stride (floats) -> half-wave B reads hit disjoint banks

typedef float v2f __attribute__((ext_vector_type(2)));
typedef float v8f __attribute__((ext_vector_type(8)));

// ---------------------------------------------------------------------------
// Kernel 1: swizzle weight[COUT][CIN][3][3] into per-lane A-fragment order.
// A-fragment (V_WMMA_F32_16X16X4_F32, wave32):
//   lanes 0-15 : VGPR0 = K0, VGPR1 = K1   (M = lane)
//   lanes 16-31: VGPR0 = K2, VGPR1 = K3   (M = lane-16)
// AW index = ((k*16 + kk)*4 + mt)*64 + lane*2 + j
// ---------------------------------------------------------------------------
__global__ void dcn_pack_weights(const float* __restrict__ w, float* __restrict__ aw) {
    int t = blockIdx.x * blockDim.x + threadIdx.x;
    if (t >= K2_ * 16 * 4 * 64) return;           // 36864 floats = 147 KB
    int j    =  t        & 1;
    int lane = (t >> 1)  & 31;
    int mt   = (t >> 6)  & 3;
    int kk   = (t >> 8)  & 15;
    int k    =  t >> 12;                           // 0..8
    int half = lane >> 4;
    int l    = lane & 15;
    int o    = mt * 16 + l;                        // output channel (M row)
    int c    = kk * 4 + half * 2 + j;              // input channel (K index)
    aw[t] = w[(o * CIN + c) * K2_ + k];
}

// ---------------------------------------------------------------------------
// Kernel 2: fused bilinear-sample + modulate + WMMA GEMM.
// One workgroup = one (n, ho) output row: D[64 x 128].
// ---------------------------------------------------------------------------
__global__ __launch_bounds__(256) void dcn_fused(
    const float* __restrict__ x,       // [N][CIN][H][W]
    const float* __restrict__ offset,  // [N][2*K2][HO][WO]
    const float* __restrict__ mask,    // [N][K2][HO][WO]
    const float* __restrict__ aw,      // packed A fragments
    float* __restrict__ out)           // [N][COUT][HO][WO]
{
    __shared__ float sS[CIN][SROW];        // 36 KB  sampled tile (one tap)
    __shared__ int   sIdx[K2_][WO_];       // 4.5 KB packed clamped corner coords
    __shared__ float sWgt[4][K2_][WO_];    // 18 KB  mask*validity*bilinear weights

    const int tid = threadIdx.x;
    const int n   = blockIdx.x / HO_;
    const int ho  = blockIdx.x % HO_;

    // ---- Phase A: per-(tap,pixel) bilinear coefficients for this row ----
    for (int e = tid; e < K2_ * WO_; e += 256) {
        int k = e / WO_, p = e - k * WO_;
        int ki = k / 3, kj = k - ki * 3;
        float dy = offset[((size_t)(n * 2 * K2_ + 2 * k    ) * HO_ + ho) * WO_ + p];
        float dx = offset[((size_t)(n * 2 * K2_ + 2 * k + 1) * HO_ + ho) * WO_ + p];
        float m  = mask  [((size_t)(n * K2_     + k        ) * HO_ + ho) * WO_ + p];
        float yf = (float)(ho - 1 + ki) + dy;
        float xf = (float)(p  - 1 + kj) + dx;
        float y0f = floorf(yf), x0f = floorf(xf);
        float wy = yf - y0f,    wx = xf - x0f;
        int y0 = (int)y0f, x0 = (int)x0f;
        int y1 = y0 + 1,   x1 = x0 + 1;
        bool vy0 = (y0 >= 0) && (y0 < HH);
        bool vy1 = (y1 >= 0) && (y1 < HH);
        bool vx0 = (x0 >= 0) && (x0 < WW);
        bool vx1 = (x1 >= 0) && (x1 < WW);
        int y0c = min(max(y0, 0), HH - 1), y1c = min(max(y1, 0), HH - 1);
        int x0c = min(max(x0, 0), WW - 1), x1c = min(max(x1, 0), WW - 1);
        sIdx[k][p] = y0c | (x0c << 8) | (y1c << 16) | (x1c << 24);
        sWgt[0][k][p] = (vy0 && vx0) ? (1.f - wy) * (1.f - wx) * m : 0.f;
        sWgt[1][k][p] = (vy0 && vx1) ? (1.f - wy) * wx          * m : 0.f;
        sWgt[2][k][p] = (vy1 && vx0) ? wy          * (1.f - wx) * m : 0.f;
        sWgt[3][k][p] = (vy1 && vx1) ? wy          * wx          * m : 0.f;
    }
    __syncthreads();

    const int wave  = tid >> 5;
    const int lane  = tid & 31;
    const int mt    = wave >> 1;          // M tile (0..3): rows mt*16..mt*16+15
    const int pbase = (wave & 1) * 64;    // column half owned by this wave
    const int half  = lane >> 4;
    const int l     = lane & 15;

    v8f acc[4] = {};                       // four 16x16 fp32 accumulator tiles
    const float* xn = x + (size_t)n * CIN * HH * WW;

    for (int k = 0; k < K2_; ++k) {
        // ---- Phase B1: build sampled tile S[c][p] for tap k (all 256 threads) ----
        for (int i = 0; i < (CIN * WO_) / 256; ++i) {   // 32 iterations
            int e = tid + i * 256;
            int c = e >> 7;          // /128
            int p = e & 127;
            int pk = sIdx[k][p];
            int y0c =  pk        & 255;
            int x0c = (pk >> 8)  & 255;
            int y1c = (pk >> 16) & 255;
            int x1c = (pk >> 24) & 255;
            const float* xc = xn + (size_t)c * HH * WW;
            float v00 = xc[y0c * WW + x0c];
            float v01 = xc[y0c * WW + x1c];
            float v10 = xc[y1c * WW + x0c];
            float v11 = xc[y1c * WW + x1c];
            sS[c][p] = v00 * sWgt[0][k][p] + v01 * sWgt[1][k][p]
                     + v10 * sWgt[2][k][p] + v11 * sWgt[3][k][p];
        }
        __syncthreads();

        // ---- Phase B2: WMMA over this tap's 64-deep K chunk ----
        const float* awk = aw + (size_t)k * 16 * 4 * 64;
        for (int kk = 0; kk < 16; ++kk) {
            v2f a = *(const v2f*)(awk + (kk * 4 + mt) * 64 + lane * 2);
            int c0 = kk * 4 + half * 2;   // B frag: lanes 0-15 -> K0/K1, 16-31 -> K2/K3
            #pragma unroll
            for (int t = 0; t < 4; ++t) {
                v2f b;
                b.x = sS[c0    ][pbase + t * 16 + l];
                b.y = sS[c0 + 1][pbase + t * 16 + l];
                acc[t] = __builtin_amdgcn_wmma_f32_16x16x4_f32(
                    /*neg_a=*/false, a, /*neg_b=*/false, b,
                    /*c_mod=*/(short)0, acc[t],
                    /*reuse_a=*/false, /*reuse_b=*/false);
            }
        }
        __syncthreads();
    }

    // ---- Phase C: write D tiles (VGPR v -> M = v + 8*half, N = l) ----
    #pragma unroll
    for (int t = 0; t < 4; ++t) {
        #pragma unroll
        for (int v = 0; v < 8; ++v) {
            int o  = mt * 16 + half * 8 + v;
            int wo = pbase + t * 16 + l;
            out[(((size_t)n * COUT + o) * HO_ + ho) * WO_ + wo] = acc[t][v];
        }
    }
}

// ---------------------------------------------------------------------------
extern "C" void kernel_launch(void* const* d_in, const int* in_sizes, int n_in,
                              void* d_out, int out_size, void* d_ws, size_t ws_size,
                              hipStream_t stream) {
    const float* x      = (const float*)d_in[0];
    const float* offset = (const float*)d_in[1];
    const float* mask   = (const float*)d_in[2];
    const float* weight = (const float*)d_in[3];
    float* out = (float*)d_out;
    float* aw  = (float*)d_ws;            // 147456 B of scratch for packed weights

    const int packElems = K2_ * 16 * 4 * 64;     // 36864
    dcn_pack_weights<<<(packElems + 255) / 256, 256, 0, stream>>>(weight, aw);
    dcn_fused<<<NB * HO_, 256, 0, stream>>>(x, offset, mask, aw, out);
}